// NCEAverage_5643587027399
// MI455X (gfx1250) — compile-verified
//
#include <hip/hip_runtime.h>
#include <math.h>

// Problem constants (match reference)
constexpr int DIM   = 128;     // feature dim D
constexpr int KK    = 4096;    // K+1
constexpr int BB    = 128;     // batch
constexpr long NB   = 500000;  // bank rows

typedef __attribute__((ext_vector_type(2))) float v2f;
typedef __attribute__((ext_vector_type(8))) float v8f;

// ---------------------------------------------------------------------------
// Kernel 1: gather + dual batched dot via V_WMMA_F32_16X16X4_F32.
// One wave (32 lanes) owns a 16-wide k-tile of one batch row b.
// A-frag (16x4 f32): lane L -> row M=L&15; VGPR pair = K{0,1} (lanes 0-15)
//                    or K{2,3} (lanes 16-31)  => one b64 load per chunk.
// B-frag (4x16 f32): every column = x[b]; same half-lane K split as A.
// D (16x16 f32): all 16 columns identical; column N=0 is lanes 0 & 16.
// ---------------------------------------------------------------------------
__global__ __launch_bounds__(256) void dot_wmma_kernel(
    const float* __restrict__ x1, const float* __restrict__ x2,
    const int*   __restrict__ idx,
    const float* __restrict__ mem1, const float* __restrict__ mem2,
    float* __restrict__ out1, float* __restrict__ out2)
{
    __shared__ float sx1[DIM];
    __shared__ float sx2[DIM];

    const int tid  = threadIdx.x;
    const int wave = tid >> 5;
    const int lane = tid & 31;
    const int tile = blockIdx.x * 8 + wave;   // global 16-row k-tile id
    const int b    = tile >> 8;               // 256 tiles per batch row; blocks never straddle b
    const int k0   = (tile & 255) << 4;

    if (tid < DIM) {                           // stage x-vectors for this b
        sx1[tid] = x1[b * DIM + tid];
        sx2[tid] = x2[b * DIM + tid];
    }
    __syncthreads();

    const int m    = lane & 15;
    const int half = lane >> 4;               // 0 -> K{0,1}, 1 -> K{2,3}
    const long row = idx[b * KK + k0 + m];
    const float* __restrict__ r2 = mem2 + row * DIM;  // out1 = w2 . x1
    const float* __restrict__ r1 = mem1 + row * DIM;  // out2 = w1 . x2

    v8f acc1 = {};
    v8f acc2 = {};
    #pragma unroll
    for (int d0 = 0; d0 < DIM; d0 += 4) {
        const int dd = d0 + 2 * half;
        v2f a1 = *(const v2f*)(r2 + dd);      // gathered A fragments
        v2f a2 = *(const v2f*)(r1 + dd);
        v2f bx1; bx1[0] = sx1[dd]; bx1[1] = sx1[dd + 1];
        v2f bx2; bx2[0] = sx2[dd]; bx2[1] = sx2[dd + 1];
        acc1 = __builtin_amdgcn_wmma_f32_16x16x4_f32(
                   false, a1, false, bx1, (short)0, acc1, false, false);
        acc2 = __builtin_amdgcn_wmma_f32_16x16x4_f32(
                   false, a2, false, bx2, (short)0, acc2, false, false);
    }

    if (m == 0) {                              // lanes 0 (M=0..7) and 16 (M=8..15)
        const int kout = b * KK + k0 + half * 8;
        #pragma unroll
        for (int r = 0; r < 8; ++r) {
            out1[kout + r] = acc1[r];
            out2[kout + r] = acc2[r];
        }
    }
}

// ---------------------------------------------------------------------------
// Kernel 2: in-place normalization. One 256-thread block per batch row.
//   p1 = softmax(out1);  m = max(p1) = 1/sum(exp(out1-max1)) exactly;
//   p2 = exp(out2 - m) / sum(exp(out2 - m))   (faithful: no max-shift on out2)
// ---------------------------------------------------------------------------
__global__ __launch_bounds__(256) void normalize_kernel(
    float* __restrict__ p1, float* __restrict__ p2)
{
    __shared__ float red[256];
    const int b = blockIdx.x, tid = threadIdx.x;
    float* o1 = p1 + (size_t)b * KK;
    float* o2 = p2 + (size_t)b * KK;

    float mx = -INFINITY;
    for (int k = tid; k < KK; k += 256) mx = fmaxf(mx, o1[k]);
    red[tid] = mx; __syncthreads();
    for (int s = 128; s > 0; s >>= 1) {
        if (tid < s) red[tid] = fmaxf(red[tid], red[tid + s]);
        __syncthreads();
    }
    const float max1 = red[0]; __syncthreads();

    float s1 = 0.f;
    for (int k = tid; k < KK; k += 256) s1 += __expf(o1[k] - max1);
    red[tid] = s1; __syncthreads();
    for (int s = 128; s > 0; s >>= 1) {
        if (tid < s) red[tid] += red[tid + s];
        __syncthreads();
    }
    const float inv1 = 1.0f / red[0]; __syncthreads();

    for (int k = tid; k < KK; k += 256) o1[k] = __expf(o1[k] - max1) * inv1;

    const float mval = inv1;                   // max(p1) == exp(0)*inv1
    float s2 = 0.f;
    for (int k = tid; k < KK; k += 256) s2 += __expf(o2[k] - mval);
    red[tid] = s2; __syncthreads();
    for (int s = 128; s > 0; s >>= 1) {
        if (tid < s) red[tid] += red[tid + s];
        __syncthreads();
    }
    const float inv2 = 1.0f / red[0]; __syncthreads();

    for (int k = tid; k < KK; k += 256) o2[k] = __expf(o2[k] - mval) * inv2;
}

// ---------------------------------------------------------------------------
// Kernel 3: stream-copy a bank to its output slot (float4, grid-stride).
// ---------------------------------------------------------------------------
__global__ __launch_bounds__(256) void copy_bank_kernel(
    const float4* __restrict__ src, float4* __restrict__ dst, long n4)
{
    long i = (long)blockIdx.x * blockDim.x + threadIdx.x;
    const long stride = (long)gridDim.x * blockDim.x;
    for (; i < n4; i += stride) dst[i] = src[i];
}

// ---------------------------------------------------------------------------
// Kernel 4: EMA + L2-normalize the 128 touched rows (after the copy).
// One 128-thread block per batch element; LDS tree-reduce the squared norm.
// ---------------------------------------------------------------------------
__global__ __launch_bounds__(128) void ema_update_kernel(
    const float* __restrict__ mem, const float* __restrict__ x,
    const int* __restrict__ index, float* __restrict__ newmem)
{
    __shared__ float red[128];
    const int b = blockIdx.x, d = threadIdx.x;
    const long r = index[b];
    const float pos = mem[r * DIM + d] * 0.5f + x[(long)b * DIM + d] * 0.5f;
    red[d] = pos * pos; __syncthreads();
    for (int s = 64; s > 0; s >>= 1) {
        if (d < s) red[d] += red[d + s];
        __syncthreads();
    }
    newmem[r * DIM + d] = pos * rsqrtf(red[0]);
}

extern "C" void kernel_launch(void* const* d_in, const int* in_sizes, int n_in,
                              void* d_out, int out_size, void* d_ws, size_t ws_size,
                              hipStream_t stream)
{
    const float* x1    = (const float*)d_in[0];
    const float* x2    = (const float*)d_in[1];
    const int*   index = (const int*)  d_in[2];
    const int*   idx   = (const int*)  d_in[3];
    const float* mem1  = (const float*)d_in[4];
    const float* mem2  = (const float*)d_in[5];

    float* out = (float*)d_out;
    float* p1  = out;
    float* p2  = p1 + (size_t)BB * KK;
    float* nm1 = p2 + (size_t)BB * KK;
    float* nm2 = nm1 + (size_t)NB * DIM;

    // 1) gather + dual GEMV via f32 WMMA; results staged into p1/p2 regions
    const int tiles = BB * (KK / 16);                 // 32768 waves
    dot_wmma_kernel<<<tiles / 8, 256, 0, stream>>>(x1, x2, idx, mem1, mem2, p1, p2);

    // 2) in-place normalization
    normalize_kernel<<<BB, 256, 0, stream>>>(p1, p2);

    // 3) copy banks to output (dominant traffic: 2 x 256 MB each way)
    const long n4 = NB * DIM / 4;
    copy_bank_kernel<<<8192, 256, 0, stream>>>((const float4*)mem1, (float4*)nm1, n4);
    copy_bank_kernel<<<8192, 256, 0, stream>>>((const float4*)mem2, (float4*)nm2, n4);

    // 4) EMA + renorm of touched rows (stream-ordered after copies)
    ema_update_kernel<<<BB, 128, 0, stream>>>(mem1, x1, index, nm1);
    ema_update_kernel<<<BB, 128, 0, stream>>>(mem2, x2, index, nm2);
}